// Corr_7490422964850
// MI455X (gfx1250) — compile-verified
//
#include <hip/hip_runtime.h>
#include <hip/hip_bf16.h>

// Shapes from the reference
#define NB     4
#define CIN    32
#define NCLS   2
#define HW     4096          // 64*64

typedef __attribute__((ext_vector_type(2))) float v2f;
typedef __attribute__((ext_vector_type(8))) float v8f;

// ---------------------------------------------------------------------------
// Kernel 1: per-pixel 1x1 conv projections  f1 = w1*x + b1, f2 = w2*x + b2
// feat: [n][32][4096], f1/f2: [n][2][4096]
// ---------------------------------------------------------------------------
__global__ void proj_kernel(const float* __restrict__ feat,
                            const float* __restrict__ w1,
                            const float* __restrict__ b1,
                            const float* __restrict__ w2,
                            const float* __restrict__ b2,
                            float* __restrict__ f1,
                            float* __restrict__ f2) {
    int t = blockIdx.x * blockDim.x + threadIdx.x;   // 0 .. NB*HW-1
    int n = t >> 12;
    int p = t & (HW - 1);
    const float* fp = feat + (size_t)n * CIN * HW + p;

    float a0 = b1[0], a1 = b1[1];
    float c0 = b2[0], c1 = b2[1];
#pragma unroll
    for (int k = 0; k < CIN; ++k) {
        float x = fp[k * HW];
        a0 = fmaf(w1[k],        x, a0);
        a1 = fmaf(w1[CIN + k],  x, a1);
        c0 = fmaf(w2[k],        x, c0);
        c1 = fmaf(w2[CIN + k],  x, c1);
    }
    f1[(n * NCLS + 0) * HW + p] = a0;
    f1[(n * NCLS + 1) * HW + p] = a1;
    f2[(n * NCLS + 0) * HW + p] = c0;
    f2[(n * NCLS + 1) * HW + p] = c1;
}

// ---------------------------------------------------------------------------
// Kernel 2: fused flash-attention, software-pipelined.
//   S^T tile (16q x 16p) = f2_tile(16x4) x f1_tile(4x16) via v_wmma_f32_16x16x4_f32
//   online softmax over q (in-lane), value accumulation for the 2 channels,
//   LDS operands for tile i+1 prefetched while tile i's exp/FMA chain runs,
//   final xor-16 half-wave merge.
// Block: 256 threads = 8 waves; wave w handles p-tile (blockIdx.x*8 + w).
// Grid: (32, NB).
// ---------------------------------------------------------------------------
__global__ void attn_kernel(const float* __restrict__ f1,
                            const float* __restrict__ f2,
                            const float* __restrict__ outv,
                            float* __restrict__ o) {
    __shared__ float2 f2s[HW + 16];    // (f2_c0[q], f2_c1[q])   + 1 pad tile
    __shared__ float2 outs[HW + 16];   // (out_c0[q], out_c1[q]) + 1 pad tile

    const int n = blockIdx.y;

    // Cooperative LDS staging (coalesced global, b64 LDS stores)
    for (int q = threadIdx.x; q < HW; q += 256) {
        float2 v;
        v.x = f2[(n * NCLS + 0) * HW + q];
        v.y = f2[(n * NCLS + 1) * HW + q];
        f2s[q] = v;
        float2 u;
        u.x = outv[(n * NCLS + 0) * HW + q];
        u.y = outv[(n * NCLS + 1) * HW + q];
        outs[q] = u;
    }
    if (threadIdx.x < 16) {            // zero the prefetch pad tile
        float2 z; z.x = 0.0f; z.y = 0.0f;
        f2s[HW + threadIdx.x] = z;
        outs[HW + threadIdx.x] = z;
    }
    __syncthreads();

    const int lane  = threadIdx.x & 31;
    const int wave  = threadIdx.x >> 5;
    const int p0    = (blockIdx.x * 8 + wave) * 16;
    const int pcol  = p0 + (lane & 15);

    // B operand = f1 columns (loop-invariant), pre-scaled by 1/sqrt(2).
    // Upper-half lanes hold K=2,3 rows -> zero them so the K padding is inert.
    const float scale = 0.70710678118654752f;
    v2f bvec;
    bvec[0] = f1[(n * NCLS + 0) * HW + pcol] * scale;
    bvec[1] = f1[(n * NCLS + 1) * HW + pcol] * scale;
    if (lane >= 16) { bvec[0] = 0.0f; bvec[1] = 0.0f; }

    const int qlane = lane & 15;          // A-row / q within tile (M index)
    const int qhalf = (lane >> 4) * 8;    // D rows: M = j (+8 for upper half)

    float m = -1.0e30f;                   // running max
    float l = 0.0f;                       // running sum(exp)
    float acc0 = 0.0f, acc1 = 0.0f;       // running sum(exp * out_c)

    // Prologue: preload tile-0 operands
    float2 a_cur = f2s[qlane];
    float2 ov_cur[8];
#pragma unroll
    for (int j = 0; j < 8; ++j) ov_cur[j] = outs[qhalf + j];

#pragma unroll 2
    for (int q0 = 0; q0 < HW; q0 += 16) {
        v2f avec; avec[0] = a_cur.x; avec[1] = a_cur.y;
        v8f cz = {};
        v8f T = __builtin_amdgcn_wmma_f32_16x16x4_f32(
            /*neg_a=*/false, avec, /*neg_b=*/false, bvec,
            /*c_mod=*/(short)0, cz, /*reuse_a=*/false, /*reuse_b=*/false);

        // Prefetch next tile's operands (pad tile makes this branch-free);
        // these only depend on q0, so they overlap the exp/FMA chain below.
        float2 a_nxt = f2s[q0 + 16 + qlane];
        float2 ov_nxt[8];
#pragma unroll
        for (int j = 0; j < 8; ++j) ov_nxt[j] = outs[q0 + 16 + qhalf + j];

        // Tile max over the 8 in-lane q rows
        float tm = T[0];
#pragma unroll
        for (int j = 1; j < 8; ++j) tm = fmaxf(tm, T[j]);

        float mn   = fmaxf(m, tm);
        float corr = __expf(m - mn);
        l    *= corr;
        acc0 *= corr;
        acc1 *= corr;
        m = mn;

#pragma unroll
        for (int j = 0; j < 8; ++j) {
            float e = __expf(T[j] - mn);
            l += e;
            acc0 = fmaf(e, ov_cur[j].x, acc0);
            acc1 = fmaf(e, ov_cur[j].y, acc1);
        }

        a_cur = a_nxt;
#pragma unroll
        for (int j = 0; j < 8; ++j) ov_cur[j] = ov_nxt[j];
    }

    // Merge the two half-wave partials (lane L and L+16 share the same p col)
    float m2  = __shfl_xor(m,    16, 32);
    float l2  = __shfl_xor(l,    16, 32);
    float a02 = __shfl_xor(acc0, 16, 32);
    float a12 = __shfl_xor(acc1, 16, 32);

    float M  = fmaxf(m, m2);
    float s1 = __expf(m  - M);
    float s2 = __expf(m2 - M);
    float L  = fmaf(l, s1, l2 * s2);
    float o0 = fmaf(acc0, s1, a02 * s2) / L;
    float o1 = fmaf(acc1, s1, a12 * s2) / L;

    if (lane < 16) {
        o[(n * NCLS + 0) * HW + pcol] = o0;
        o[(n * NCLS + 1) * HW + pcol] = o1;
    }
}

// ---------------------------------------------------------------------------
extern "C" void kernel_launch(void* const* d_in, const int* in_sizes, int n_in,
                              void* d_out, int out_size, void* d_ws, size_t ws_size,
                              hipStream_t stream) {
    const float* feat = (const float*)d_in[0];   // [4,32,64,64]
    const float* outv = (const float*)d_in[1];   // [4,2,64,64]
    const float* w1   = (const float*)d_in[2];   // [2,32]
    const float* b1   = (const float*)d_in[3];   // [2]
    const float* w2   = (const float*)d_in[4];   // [2,32]
    const float* b2   = (const float*)d_in[5];   // [2]
    float*       out  = (float*)d_out;           // [4,2,64,64]

    float* f1 = (float*)d_ws;                    // [4,2,4096]
    float* f2 = f1 + NB * NCLS * HW;             // [4,2,4096]  (256 KB total)

    proj_kernel<<<(NB * HW) / 256, 256, 0, stream>>>(feat, w1, b1, w2, b2, f1, f2);

    dim3 grid(HW / 128, NB);                     // 32 p-blocks x 4 batches
    attn_kernel<<<grid, 256, 0, stream>>>(f1, f2, outv, out);
}